// LearnedWLSEmbeddings_19937238188037
// MI455X (gfx1250) — compile-verified
//
#include <hip/hip_runtime.h>
#include <stdint.h>

// LearnedWLSEmbeddings fused kernel for MI455X (gfx1250, wave32).
// One wave = one batch (M=32 points = 32 lanes). fp32 throughout.
// The A-net output layer (32x4 @ 4x64) runs on V_WMMA_F32_16X16X4_F32.

#define BATCHES 32768
#define MPTS    32
#define EDIM    64
#define WAVES   4           // waves (=batches) per block
#define NS      65          // padded LDS row stride for n[m][e] (bank-conflict-free)

typedef float v2f __attribute__((ext_vector_type(2)));
typedef float v8f __attribute__((ext_vector_type(8)));

struct Params {
    const float*   x;       // (B,32,4)
    const uint8_t* pad;     // (B,32) bool
    const uint8_t* maskb;   // (B,)   bool
    const float *WA0,*bA0,*WA1,*bA1,*WA2,*bA2,*WA3,*bA3;   // A_params
    const float *Wb0,*bb0,*Wb1,*bb1,*Wb2,*bb2,*Wb3,*bb3;   // b_params
    const float *WD0,*bD0,*WD1,*bD1,*WD2,*bD2;             // dec_params
    const float* Z;         // (1,64)
    float*       out;       // (B,4)
};

__device__ __forceinline__ float wave_sum32(float v) {
    v += __shfl_xor(v, 1, 32);
    v += __shfl_xor(v, 2, 32);
    v += __shfl_xor(v, 4, 32);
    v += __shfl_xor(v, 8, 32);
    v += __shfl_xor(v, 16, 32);
    return v;
}

// General 4x4 inverse by adjugate (wave-uniform inputs -> uniform result).
__device__ __forceinline__ void inv4(const float m[16], float o[16]) {
    float inv[16];
    inv[0]  =  m[5]*m[10]*m[15] - m[5]*m[11]*m[14] - m[9]*m[6]*m[15] + m[9]*m[7]*m[14] + m[13]*m[6]*m[11] - m[13]*m[7]*m[10];
    inv[4]  = -m[4]*m[10]*m[15] + m[4]*m[11]*m[14] + m[8]*m[6]*m[15] - m[8]*m[7]*m[14] - m[12]*m[6]*m[11] + m[12]*m[7]*m[10];
    inv[8]  =  m[4]*m[9]*m[15]  - m[4]*m[11]*m[13] - m[8]*m[5]*m[15] + m[8]*m[7]*m[13] + m[12]*m[5]*m[11] - m[12]*m[7]*m[9];
    inv[12] = -m[4]*m[9]*m[14]  + m[4]*m[10]*m[13] + m[8]*m[5]*m[14] - m[8]*m[6]*m[13] - m[12]*m[5]*m[10] + m[12]*m[6]*m[9];
    inv[1]  = -m[1]*m[10]*m[15] + m[1]*m[11]*m[14] + m[9]*m[2]*m[15] - m[9]*m[3]*m[14] - m[13]*m[2]*m[11] + m[13]*m[3]*m[10];
    inv[5]  =  m[0]*m[10]*m[15] - m[0]*m[11]*m[14] - m[8]*m[2]*m[15] + m[8]*m[3]*m[14] + m[12]*m[2]*m[11] - m[12]*m[3]*m[10];
    inv[9]  = -m[0]*m[9]*m[15]  + m[0]*m[11]*m[13] + m[8]*m[1]*m[15] - m[8]*m[3]*m[13] - m[12]*m[1]*m[11] + m[12]*m[3]*m[9];
    inv[13] =  m[0]*m[9]*m[14]  - m[0]*m[10]*m[13] - m[8]*m[1]*m[14] + m[8]*m[2]*m[13] + m[12]*m[1]*m[10] - m[12]*m[2]*m[9];
    inv[2]  =  m[1]*m[6]*m[15]  - m[1]*m[7]*m[14]  - m[5]*m[2]*m[15] + m[5]*m[3]*m[14] + m[13]*m[2]*m[7]  - m[13]*m[3]*m[6];
    inv[6]  = -m[0]*m[6]*m[15]  + m[0]*m[7]*m[14]  + m[4]*m[2]*m[15] - m[4]*m[3]*m[14] - m[12]*m[2]*m[7]  + m[12]*m[3]*m[6];
    inv[10] =  m[0]*m[5]*m[15]  - m[0]*m[7]*m[13]  - m[4]*m[1]*m[15] + m[4]*m[3]*m[13] + m[12]*m[1]*m[7]  - m[12]*m[3]*m[5];
    inv[14] = -m[0]*m[5]*m[14]  + m[0]*m[6]*m[13]  + m[4]*m[1]*m[14] - m[4]*m[2]*m[13] - m[12]*m[1]*m[6]  + m[12]*m[2]*m[5];
    inv[3]  = -m[1]*m[6]*m[11]  + m[1]*m[7]*m[10]  + m[5]*m[2]*m[11] - m[5]*m[3]*m[10] - m[9]*m[2]*m[7]   + m[9]*m[3]*m[6];
    inv[7]  =  m[0]*m[6]*m[11]  - m[0]*m[7]*m[10]  - m[4]*m[2]*m[11] + m[4]*m[3]*m[10] + m[8]*m[2]*m[7]   - m[8]*m[3]*m[6];
    inv[11] = -m[0]*m[5]*m[11]  + m[0]*m[7]*m[9]   + m[4]*m[1]*m[11] - m[4]*m[3]*m[9]  - m[8]*m[1]*m[7]   + m[8]*m[3]*m[5];
    inv[15] =  m[0]*m[5]*m[10]  - m[0]*m[6]*m[9]   - m[4]*m[1]*m[10] + m[4]*m[2]*m[9]  + m[8]*m[1]*m[6]   - m[8]*m[2]*m[5];
    const float det = m[0]*inv[0] + m[1]*inv[4] + m[2]*inv[8] + m[3]*inv[12];
    const float rd  = 1.0f / det;
#pragma unroll
    for (int i = 0; i < 16; ++i) o[i] = inv[i] * rd;
}

__global__ __launch_bounds__(WAVES * 32)
void wls_fused_kernel(Params p) {
    // dec-MLP weights cached per block: [WD0(1088) bD0(16) WD1(256) bD1(16) WD2(64) bD2(4)]
    __shared__ float sdec[1444];
    __shared__ float s_n [WAVES][MPTS * NS];        // n[m][e], padded stride
    __shared__ __align__(16) float s_h3[WAVES][MPTS * 4];
    __shared__ float s_sm[WAVES][MPTS];
    __shared__ float s_oE[WAVES][EDIM];
    __shared__ float s_h1[WAVES][16];
    __shared__ float s_h2[WAVES][16];

    const int tid   = threadIdx.x;
    const int w     = tid >> 5;
    const int lane  = tid & 31;
    const int batch = blockIdx.x * WAVES + w;
    const int m     = lane;

    // ---- cooperative load of dec weights into LDS ----
    for (int i = tid; i < 1088; i += WAVES*32) sdec[i]        = p.WD0[i];
    for (int i = tid; i < 16;   i += WAVES*32) sdec[1088 + i] = p.bD0[i];
    for (int i = tid; i < 256;  i += WAVES*32) sdec[1104 + i] = p.WD1[i];
    for (int i = tid; i < 16;   i += WAVES*32) sdec[1360 + i] = p.bD1[i];
    for (int i = tid; i < 64;   i += WAVES*32) sdec[1376 + i] = p.WD2[i];
    for (int i = tid; i < 4;    i += WAVES*32) sdec[1440 + i] = p.bD2[i];

    // ---- per-point features ----
    const float4 xv = *(const float4*)(p.x + ((size_t)batch * MPTS + m) * 4);
    const float padf = p.pad[(size_t)batch * MPTS + m] ? 0.0f : 1.0f;
    float xin[5] = { xv.x, xv.y, xv.z, xv.w, padf };
    const float a0 = -xv.y, a1 = -xv.z, a2 = -xv.w, a3 = 1.0f;

    // ---- A-net hidden layers (5->4->4->4), wave-uniform weights => s_loads ----
    float hA[4], tA[4];
#pragma unroll
    for (int j = 0; j < 4; ++j) {
        float acc = p.bA0[j];
#pragma unroll
        for (int i = 0; i < 5; ++i) acc = fmaf(xin[i], p.WA0[i*4 + j], acc);
        hA[j] = fmaxf(acc, 0.0f);
    }
#pragma unroll
    for (int j = 0; j < 4; ++j) {
        float acc = p.bA1[j];
#pragma unroll
        for (int i = 0; i < 4; ++i) acc = fmaf(hA[i], p.WA1[i*4 + j], acc);
        tA[j] = fmaxf(acc, 0.0f);
    }
#pragma unroll
    for (int j = 0; j < 4; ++j) {
        float acc = p.bA2[j];
#pragma unroll
        for (int i = 0; i < 4; ++i) acc = fmaf(tA[i], p.WA2[i*4 + j], acc);
        hA[j] = fmaxf(acc, 0.0f);                      // hA = h3 (input to WMMA layer)
    }

    // ---- b-net (5->4->4->4->1) ----
    float hB[4], tB[4];
#pragma unroll
    for (int j = 0; j < 4; ++j) {
        float acc = p.bb0[j];
#pragma unroll
        for (int i = 0; i < 5; ++i) acc = fmaf(xin[i], p.Wb0[i*4 + j], acc);
        hB[j] = fmaxf(acc, 0.0f);
    }
#pragma unroll
    for (int j = 0; j < 4; ++j) {
        float acc = p.bb1[j];
#pragma unroll
        for (int i = 0; i < 4; ++i) acc = fmaf(hB[i], p.Wb1[i*4 + j], acc);
        tB[j] = fmaxf(acc, 0.0f);
    }
#pragma unroll
    for (int j = 0; j < 4; ++j) {
        float acc = p.bb2[j];
#pragma unroll
        for (int i = 0; i < 4; ++i) acc = fmaf(tB[i], p.Wb2[i*4 + j], acc);
        hB[j] = fmaxf(acc, 0.0f);
    }
    float rm = xv.x + p.bb3[0];
#pragma unroll
    for (int i = 0; i < 4; ++i) rm = fmaf(hB[i], p.Wb3[i], rm);

    // ---- stage h3 for WMMA A-operand layout ----
    ((float4*)s_h3[w])[m] = make_float4(hA[0], hA[1], hA[2], hA[3]);
    __syncthreads();

    // ---- n = H3(32x4) @ WA3(4x64) + bA3 via V_WMMA_F32_16X16X4_F32 ----
    // A (16x4 f32): lanes 0-15 -> M rows, vgpr0/1 = K0/K1 (low half) or K2/K3 (high half)
    const int l16 = lane & 15;
    const int k0  = (lane < 16) ? 0 : 2;
    const int mb  = (lane < 16) ? 0 : 8;     // D-layout row offset per lane half
    v2f aop[2];
    aop[0] = *(const v2f*)&s_h3[w][(0*16 + l16) * 4 + k0];
    aop[1] = *(const v2f*)&s_h3[w][(1*16 + l16) * 4 + k0];
#pragma unroll
    for (int s = 0; s < 4; ++s) {
        const int e = s * 16 + l16;
        v2f bop;
        bop[0] = p.WA3[(k0    ) * EDIM + e];
        bop[1] = p.WA3[(k0 + 1) * EDIM + e];
        const float cb = p.bA3[e];            // bias folded into C (depends on N only)
        v8f c = { cb, cb, cb, cb, cb, cb, cb, cb };
#pragma unroll
        for (int t = 0; t < 2; ++t) {
            v8f d = __builtin_amdgcn_wmma_f32_16x16x4_f32(
                false, aop[t], false, bop, (short)0, c, false, false);
#pragma unroll
            for (int v = 0; v < 8; ++v)
                s_n[w][(t*16 + mb + v) * NS + e] = d[v];   // scatter D -> n[m][e]
        }
    }

    // ---- AtA (10 unique) and u = A_base^T r : wave butterfly reductions ----
    float red[14] = { a0*a0, a0*a1, a0*a2, a0*a3, a1*a1, a1*a2, a1*a3,
                      a2*a2, a2*a3, a3*a3, a0*rm, a1*rm, a2*rm, a3*rm };
#pragma unroll
    for (int k = 0; k < 14; ++k) red[k] = wave_sum32(red[k]);

    float AtA[16] = { red[0], red[1], red[2], red[3],
                      red[1], red[4], red[5], red[6],
                      red[2], red[5], red[7], red[8],
                      red[3], red[6], red[8], red[9] };
    float Ainv[16];
    inv4(AtA, Ainv);
    const float u0 = red[10], u1 = red[11], u2 = red[12], u3 = red[13];

    float pv[4];
#pragma unroll
    for (int k = 0; k < 4; ++k)
        pv[k] = Ainv[k*4+0]*u0 + Ainv[k*4+1]*u1 + Ainv[k*4+2]*u2 + Ainv[k*4+3]*u3;

    s_sm[w][m] = rm - (a0*pv[0] + a1*pv[1] + a2*pv[2] + a3*pv[3]);
    __syncthreads();

    // ---- pass 1: outE[e] = Z[e] * sum_m n[m][e] * s_m   (lane <-> e, 2 e's/lane) ----
    {
        const int e0 = lane, e1 = lane + 32;
        float w0 = 0.0f, w1 = 0.0f;
#pragma unroll 4
        for (int mm = 0; mm < MPTS; ++mm) {
            const float smv = s_sm[w][mm];                 // LDS broadcast
            w0 = fmaf(s_n[w][mm*NS + e0], smv, w0);
            w1 = fmaf(s_n[w][mm*NS + e1], smv, w1);
        }
        s_oE[w][e0] = p.Z[e0] * w0;
        s_oE[w][e1] = p.Z[e1] * w1;
    }
    __syncthreads();

    // ---- pass 2: t_m = n_m . outE ; q = sum_m a_m t_m ; out4 = Ainv (u - q) ----
    float tm = 0.0f;
#pragma unroll 8
    for (int e = 0; e < EDIM; ++e)
        tm = fmaf(s_n[w][m*NS + e], s_oE[w][e], tm);

    const float q0 = wave_sum32(a0 * tm);
    const float q1 = wave_sum32(a1 * tm);
    const float q2 = wave_sum32(a2 * tm);
    const float q3 = wave_sum32(a3 * tm);

    float o4[4];
    {
        const float d0 = u0 - q0, d1 = u1 - q1, d2 = u2 - q2, d3 = u3 - q3;
#pragma unroll
        for (int k = 0; k < 4; ++k)
            o4[k] = Ainv[k*4+0]*d0 + Ainv[k*4+1]*d1 + Ainv[k*4+2]*d2 + Ainv[k*4+3]*d3;
    }

    // ---- dec MLP: in = [o4(4), outE(64)] -> 16 -> 16 -> 4 ----
    {
        const int j = l16;                                 // both halves redundant
        float acc = sdec[1088 + j];
#pragma unroll
        for (int i = 0; i < 4; ++i) acc = fmaf(o4[i], sdec[i*16 + j], acc);
#pragma unroll 8
        for (int i = 0; i < EDIM; ++i) acc = fmaf(sdec[(4 + i)*16 + j], s_oE[w][i], acc);
        if (lane < 16) s_h1[w][j] = fmaxf(acc, 0.0f);
    }
    __syncthreads();
    {
        const int j = l16;
        float acc = sdec[1360 + j];
#pragma unroll
        for (int i = 0; i < 16; ++i) acc = fmaf(sdec[1104 + i*16 + j], s_h1[w][i], acc);
        if (lane < 16) s_h2[w][j] = fmaxf(acc, 0.0f);
    }
    __syncthreads();
    {
        const int k = lane & 3;
        float acc = sdec[1440 + k];
#pragma unroll
        for (int i = 0; i < 16; ++i) acc = fmaf(sdec[1376 + i*4 + k], s_h2[w][i], acc);
        const float maskf = p.maskb[batch] ? 1.0f : 0.0f;
        const float outv  = (o4[k] + acc) * maskf;
        if (lane < 4) p.out[(size_t)batch * 4 + k] = outv;
    }
}

extern "C" void kernel_launch(void* const* d_in, const int* in_sizes, int n_in,
                              void* d_out, int out_size, void* d_ws, size_t ws_size,
                              hipStream_t stream) {
    (void)in_sizes; (void)n_in; (void)out_size; (void)d_ws; (void)ws_size;
    Params P;
    P.x     = (const float*)d_in[0];
    P.pad   = (const uint8_t*)d_in[1];
    P.maskb = (const uint8_t*)d_in[2];
    P.WA0 = (const float*)d_in[3];  P.bA0 = (const float*)d_in[4];
    P.WA1 = (const float*)d_in[5];  P.bA1 = (const float*)d_in[6];
    P.WA2 = (const float*)d_in[7];  P.bA2 = (const float*)d_in[8];
    P.WA3 = (const float*)d_in[9];  P.bA3 = (const float*)d_in[10];
    P.Wb0 = (const float*)d_in[11]; P.bb0 = (const float*)d_in[12];
    P.Wb1 = (const float*)d_in[13]; P.bb1 = (const float*)d_in[14];
    P.Wb2 = (const float*)d_in[15]; P.bb2 = (const float*)d_in[16];
    P.Wb3 = (const float*)d_in[17]; P.bb3 = (const float*)d_in[18];
    P.WD0 = (const float*)d_in[19]; P.bD0 = (const float*)d_in[20];
    P.WD1 = (const float*)d_in[21]; P.bD1 = (const float*)d_in[22];
    P.WD2 = (const float*)d_in[23]; P.bD2 = (const float*)d_in[24];
    P.Z   = (const float*)d_in[25];
    P.out = (float*)d_out;

    dim3 grid(BATCHES / WAVES);
    dim3 block(WAVES * 32);
    hipLaunchKernelGGL(wls_fused_kernel, grid, block, 0, stream, P);
}